// GAT_V2_2748779070164
// MI455X (gfx1250) — compile-verified
//
#include <hip/hip_runtime.h>
#include <cmath>

typedef __attribute__((ext_vector_type(2))) float v2f;
typedef __attribute__((ext_vector_type(8))) float v8f;

constexpr int N_NODES = 10000;
constexpr int N_EDGES = 160000;
constexpr int E_TOT   = N_EDGES + N_NODES;   // 170000 (self-loops appended)
constexpr int F1 = 960, H1 = 8, C1 = 120;
constexpr float NEG_SLOPE = 0.2f;

// ---------------------------------------------------------------- helpers
__device__ __forceinline__ void edge_endpoints(const int* __restrict__ ei,
                                               int e, int& s, int& d) {
  if (e < N_EDGES) { s = ei[e]; d = ei[N_EDGES + e]; }
  else             { s = e - N_EDGES; d = s; }        // self loop
}

__device__ __forceinline__ void atomic_max_f32(float* addr, float val) {
  int* ia = (int*)addr;
  int old = __float_as_int(*addr);
  while (__int_as_float(old) < val) {
    int assumed = old;
    old = atomicCAS(ia, assumed, __float_as_int(val));
    if (old == assumed) break;
  }
}

// ---------------------------------------------------------------- init
// zero h1-accumulator & denominators, -inf maxima, seed out with bias2
__global__ void k_init(float* __restrict__ h1, float* __restrict__ max1,
                       float* __restrict__ den1, float* __restrict__ max2,
                       float* __restrict__ den2, float* __restrict__ out,
                       const float* __restrict__ bias2) {
  int idx = blockIdx.x * blockDim.x + threadIdx.x;
  if (idx < N_NODES * F1) h1[idx] = 0.f;
  if (idx < N_NODES * H1) { max1[idx] = -INFINITY; den1[idx] = 0.f; }
  if (idx < N_NODES)      { max2[idx] = -INFINITY; den2[idx] = 0.f; }
  if (idx < N_NODES * 4)  out[idx] = bias2[idx & 3];
}

// ---------------------------------------------------------------- layer-1 GEMM
// out[16x16 tile] = x[16x4] * W[4x16] + b ; one V_WMMA_F32_16X16X4_F32 per wave
__global__ __launch_bounds__(32)
void k_gemm1(const float* __restrict__ x, const float* __restrict__ W,
             const float* __restrict__ b, float* __restrict__ out) {
  const int tn = blockIdx.x;            // 0..59   (960/16)
  const int tm = blockIdx.y;            // 0..624  (10000/16)
  const int lane = threadIdx.x;
  const int koff = (lane >> 4) << 1;    // lanes 0-15: K=0,1 ; 16-31: K=2,3
  const int rowA = tm * 16 + (lane & 15);
  const int colB = tn * 16 + (lane & 15);

  v2f a, bm;
  a.x  = x[rowA * 4 + koff];
  a.y  = x[rowA * 4 + koff + 1];
  bm.x = W[koff * F1 + colB];
  bm.y = W[(koff + 1) * F1 + colB];

  const float bb = b[colB];
  v8f c;
  #pragma unroll
  for (int v = 0; v < 8; ++v) c[v] = bb;

  c = __builtin_amdgcn_wmma_f32_16x16x4_f32(false, a, false, bm,
                                            (short)0, c, false, false);

  const int m0 = (lane >> 4) * 8;       // C/D: lanes 16-31 hold M=8..15
  const int n  = lane & 15;
  #pragma unroll
  for (int v = 0; v < 8; ++v)
    out[(tm * 16 + m0 + v) * F1 + tn * 16 + n] = c[v];
}

// ---------------------------------------------------------------- layer-1 edge logits
__global__ void k_logits1(const int* __restrict__ ei, const float* __restrict__ xl,
                          const float* __restrict__ xr, const float* __restrict__ att,
                          float* __restrict__ elog, float* __restrict__ nmax) {
  int idx = blockIdx.x * blockDim.x + threadIdx.x;
  if (idx >= E_TOT * H1) return;
  const int e = idx >> 3, h = idx & 7;
  int s, d; edge_endpoints(ei, e, s, d);

  const float4* pl = (const float4*)(xl + s * F1 + h * C1);
  const float4* pr = (const float4*)(xr + d * F1 + h * C1);
  const float4* pa = (const float4*)(att + h * C1);
  float acc = 0.f;
  #pragma unroll 6
  for (int i = 0; i < C1 / 4; ++i) {
    float4 a4 = pa[i], l4 = pl[i], r4 = pr[i];
    float s0 = l4.x + r4.x; s0 = s0 > 0.f ? s0 : NEG_SLOPE * s0; acc += a4.x * s0;
    float s1 = l4.y + r4.y; s1 = s1 > 0.f ? s1 : NEG_SLOPE * s1; acc += a4.y * s1;
    float s2 = l4.z + r4.z; s2 = s2 > 0.f ? s2 : NEG_SLOPE * s2; acc += a4.z * s2;
    float s3 = l4.w + r4.w; s3 = s3 > 0.f ? s3 : NEG_SLOPE * s3; acc += a4.w * s3;
  }
  elog[idx] = acc;
  atomic_max_f32(&nmax[d * H1 + h], acc);
}

__global__ void k_exp1(const int* __restrict__ ei, float* __restrict__ elog,
                       const float* __restrict__ nmax, float* __restrict__ den) {
  int idx = blockIdx.x * blockDim.x + threadIdx.x;
  if (idx >= E_TOT * H1) return;
  const int e = idx >> 3, h = idx & 7;
  int s, d; edge_endpoints(ei, e, s, d);
  const float v = __expf(elog[idx] - nmax[d * H1 + h]);
  elog[idx] = v;
  atomicAdd(&den[d * H1 + h], v);
}

// one block per edge, one lane per output channel -> coalesced f32 atomics
__global__ __launch_bounds__(960)
void k_aggr1(const int* __restrict__ ei, const float* __restrict__ xl,
             const float* __restrict__ elog, const float* __restrict__ den,
             float* __restrict__ hout) {
  const int e = blockIdx.x;
  const int c = threadIdx.x;
  const int h = c / C1;
  int s, d; edge_endpoints(ei, e, s, d);
  const float alpha = elog[e * H1 + h] / (den[d * H1 + h] + 1e-16f);
  atomicAdd(&hout[d * F1 + c], alpha * xl[s * F1 + c]);
}

__global__ void k_bias_relu1(float* __restrict__ h1, const float* __restrict__ bias1) {
  int idx = blockIdx.x * blockDim.x + threadIdx.x;
  if (idx >= N_NODES * F1) return;
  const float v = h1[idx] + bias1[idx % F1];
  h1[idx] = v > 0.f ? v : 0.f;
}

// ---------------------------------------------------------------- layer-2 GEMM
// [16x960] * [960x8] via 240 chained WMMA 16x16x4; B = [W2l|W2r] staged in LDS
__global__ __launch_bounds__(32)
void k_gemm2(const float* __restrict__ h, const float* __restrict__ W2l,
             const float* __restrict__ W2r, const float* __restrict__ b2l,
             const float* __restrict__ b2r, float* __restrict__ xl2,
             float* __restrict__ xr2) {
  __shared__ float Bs[F1 * 8];
  const int lane = threadIdx.x;
  for (int i = lane; i < F1 * 8; i += 32) {
    const int k = i >> 3, c = i & 7;
    Bs[i] = (c < 4) ? W2l[k * 4 + c] : W2r[k * 4 + (c - 4)];
  }
  __syncthreads();

  const int tm   = blockIdx.x;               // 0..624
  const int rowA = tm * 16 + (lane & 15);
  const int col  = lane & 15;
  const int koff = (lane >> 4) << 1;

  v8f c;
  const float cb = (col < 4) ? b2l[col] : ((col < 8) ? b2r[col - 4] : 0.f);
  #pragma unroll
  for (int v = 0; v < 8; ++v) c[v] = cb;

  const float* hrow  = h + rowA * F1;
  const bool   valid = col < 8;
  for (int kb = 0; kb < F1; kb += 4) {
    const int k0 = kb + koff;
    v2f a, bm;
    a.x  = hrow[k0];
    a.y  = hrow[k0 + 1];
    bm.x = valid ? Bs[k0 * 8 + col] : 0.f;
    bm.y = valid ? Bs[(k0 + 1) * 8 + col] : 0.f;
    c = __builtin_amdgcn_wmma_f32_16x16x4_f32(false, a, false, bm,
                                              (short)0, c, false, false);
  }

  const int m0 = (lane >> 4) * 8;
  const int n  = lane & 15;
  #pragma unroll
  for (int v = 0; v < 8; ++v) {
    const int row = tm * 16 + m0 + v;
    if (n < 4)      xl2[row * 4 + n]       = c[v];
    else if (n < 8) xr2[row * 4 + (n - 4)] = c[v];
  }
}

// ---------------------------------------------------------------- layer-2 attention
__global__ void k_logits2(const int* __restrict__ ei, const float* __restrict__ xl2,
                          const float* __restrict__ xr2, const float* __restrict__ att2,
                          float* __restrict__ elog2, float* __restrict__ nmax2) {
  int e = blockIdx.x * blockDim.x + threadIdx.x;
  if (e >= E_TOT) return;
  int s, d; edge_endpoints(ei, e, s, d);
  float acc = 0.f;
  #pragma unroll
  for (int c = 0; c < 4; ++c) {
    float sv = xl2[s * 4 + c] + xr2[d * 4 + c];
    sv = sv > 0.f ? sv : NEG_SLOPE * sv;
    acc += att2[c] * sv;
  }
  elog2[e] = acc;
  atomic_max_f32(&nmax2[d], acc);
}

__global__ void k_exp2(const int* __restrict__ ei, float* __restrict__ elog2,
                       const float* __restrict__ nmax2, float* __restrict__ den2) {
  int e = blockIdx.x * blockDim.x + threadIdx.x;
  if (e >= E_TOT) return;
  int s, d; edge_endpoints(ei, e, s, d);
  const float v = __expf(elog2[e] - nmax2[d]);
  elog2[e] = v;
  atomicAdd(&den2[d], v);
}

__global__ void k_aggr2(const int* __restrict__ ei, const float* __restrict__ xl2,
                        const float* __restrict__ elog2, const float* __restrict__ den2,
                        float* __restrict__ out) {
  int idx = blockIdx.x * blockDim.x + threadIdx.x;
  if (idx >= E_TOT * 4) return;
  const int e = idx >> 2, c = idx & 3;
  int s, d; edge_endpoints(ei, e, s, d);
  const float alpha = elog2[e] / (den2[d] + 1e-16f);
  atomicAdd(&out[d * 4 + c], alpha * xl2[s * 4 + c]);
}

// ---------------------------------------------------------------- launcher
extern "C" void kernel_launch(void* const* d_in, const int* in_sizes, int n_in,
                              void* d_out, int out_size, void* d_ws, size_t ws_size,
                              hipStream_t stream) {
  const float* x     = (const float*)d_in[0];
  const int*   ei    = (const int*)d_in[1];
  const float* W1l   = (const float*)d_in[2];
  const float* W1r   = (const float*)d_in[3];
  const float* b1l   = (const float*)d_in[4];
  const float* b1r   = (const float*)d_in[5];
  const float* att1  = (const float*)d_in[6];
  const float* bias1 = (const float*)d_in[7];
  const float* W2l   = (const float*)d_in[8];
  const float* W2r   = (const float*)d_in[9];
  const float* b2l   = (const float*)d_in[10];
  const float* b2r   = (const float*)d_in[11];
  const float* att2  = (const float*)d_in[12];
  const float* bias2 = (const float*)d_in[13];
  float* out = (float*)d_out;

  // workspace layout (floats); total ~30.7M floats (~123 MB)
  float* ws   = (float*)d_ws;
  float* xl1  = ws;                       // N*F1 = 9,600,000
  float* xr1  = xl1  + N_NODES * F1;
  float* h1   = xr1  + N_NODES * F1;      // aggregation accumulator / layer-2 input
  float* e1   = h1   + N_NODES * F1;      // E_TOT*H1 = 1,360,000
  float* max1 = e1   + E_TOT * H1;        // 80,000
  float* den1 = max1 + N_NODES * H1;      // 80,000
  float* xl2  = den1 + N_NODES * H1;      // 40,000
  float* xr2  = xl2  + N_NODES * 4;       // 40,000
  float* e2   = xr2  + N_NODES * 4;       // 170,000
  float* max2 = e2   + E_TOT;             // 10,000
  float* den2 = max2 + N_NODES;           // 10,000

  const int B = 256;

  k_init<<<(N_NODES * F1 + B - 1) / B, B, 0, stream>>>(h1, max1, den1, max2, den2,
                                                       out, bias2);

  dim3 g1(F1 / 16, N_NODES / 16);         // 60 x 625 tiles
  k_gemm1<<<g1, 32, 0, stream>>>(x, W1l, b1l, xl1);
  k_gemm1<<<g1, 32, 0, stream>>>(x, W1r, b1r, xr1);

  k_logits1<<<(E_TOT * H1 + B - 1) / B, B, 0, stream>>>(ei, xl1, xr1, att1, e1, max1);
  k_exp1   <<<(E_TOT * H1 + B - 1) / B, B, 0, stream>>>(ei, e1, max1, den1);
  k_aggr1  <<<E_TOT, F1, 0, stream>>>(ei, xl1, e1, den1, h1);
  k_bias_relu1<<<(N_NODES * F1 + B - 1) / B, B, 0, stream>>>(h1, bias1);

  k_gemm2<<<N_NODES / 16, 32, 0, stream>>>(h1, W2l, W2r, b2l, b2r, xl2, xr2);

  k_logits2<<<(E_TOT + B - 1) / B, B, 0, stream>>>(ei, xl2, xr2, att2, e2, max2);
  k_exp2   <<<(E_TOT + B - 1) / B, B, 0, stream>>>(ei, e2, max2, den2);
  k_aggr2  <<<(E_TOT * 4 + B - 1) / B, B, 0, stream>>>(ei, xl2, e2, den2, out);
}